// BurstGNN_88484916232714
// MI455X (gfx1250) — compile-verified
//
#include <hip/hip_runtime.h>

#define EPSF 0.1f

typedef __attribute__((ext_vector_type(2))) float v2f;
typedef __attribute__((ext_vector_type(8))) float v8f;

#if defined(__has_builtin)
#if __has_builtin(__builtin_amdgcn_wmma_f32_16x16x4_f32)
#define HAVE_WMMA_F32X4 1
#endif
#endif

__device__ __forceinline__ float lrelu(float v) { return v > 0.0f ? v : 0.01f * v; }

// 16x16 f32 tile: D = A(16x64, LDS rows, stride 68) * B(64 x ldB, global, cols n0..n0+15)
// One full wave32; EXEC must be all ones. ISA layouts:
//  A 16x4 f32: lanes 0-15 row M=lane hold K=k0,k0+1; lanes 16-31 hold K=k0+2,k0+3
//  B 4x16 f32: lanes 0-15 col N=lane hold K=k0,k0+1; lanes 16-31 hold K=k0+2,k0+3
//  C/D 16x16 f32: VGPR r -> (M = r + (lane>=16 ? 8 : 0), N = lane&15)
__device__ __forceinline__ v8f wmma_tile64(const float A[16][68],
                                           const float* __restrict__ B,
                                           int ldB, int n0, int lane) {
  v8f acc = {};
#if HAVE_WMMA_F32X4
  const int half = (lane >> 4) << 1;  // +0 for lanes 0-15, +2 for lanes 16-31
  const int idx  = lane & 15;         // A row / B col within the tile
  for (int k0 = 0; k0 < 64; k0 += 4) {
    const int k = k0 + half;
    v2f a, b;
    a.x = A[idx][k];
    a.y = A[idx][k + 1];
    b.x = B[k * ldB + n0 + idx];
    b.y = B[(k + 1) * ldB + n0 + idx];
    acc = __builtin_amdgcn_wmma_f32_16x16x4_f32(false, a, false, b, (short)0,
                                                acc, false, false);
  }
#else
  const int idx = lane & 15;
  const int mb  = (lane >> 4) << 3;
  for (int r = 0; r < 8; ++r) {
    float s = 0.0f;
    for (int k = 0; k < 64; ++k) s += A[mb + r][k] * B[k * ldB + n0 + idx];
    acc[r] = s;
  }
#endif
  return acc;
}

// ---------------------------------------------------------------------------
// K1: per-node feature MLP -> x[N,64].  One wave per 16-node tile.
// ---------------------------------------------------------------------------
__global__ void __launch_bounds__(32)
node_mlp_kernel(const float* __restrict__ num_prop,   // [N,20]
                const float* __restrict__ cat_prop,   // [N,12]
                const float* __restrict__ W_num,      // [20,32]
                const float* __restrict__ b_num,      // [32]
                const float* __restrict__ W_cat,      // [12,32]
                const float* __restrict__ b_cat,      // [32]
                const float* __restrict__ W_tog,      // [64,64]
                const float* __restrict__ b_tog,      // [64]
                float* __restrict__ x, int N) {
  __shared__ float tS[16][68];
  const int lane = threadIdx.x;
  const int m0 = blockIdx.x * 16;

  // small MLPs -> concat tile t[16][64] in LDS (32 elems/lane)
  for (int e = lane; e < 16 * 64; e += 32) {
    const int m = e >> 6, j = e & 63;
    const int node = m0 + m;
    float s;
    if (j < 32) {
      s = b_num[j];
      const float* row = num_prop + node * 20;
      for (int k = 0; k < 20; ++k) s += row[k] * W_num[k * 32 + j];
    } else {
      const int jj = j - 32;
      s = b_cat[jj];
      const float* row = cat_prop + node * 12;
      for (int k = 0; k < 12; ++k) s += row[k] * W_cat[k * 32 + jj];
    }
    tS[m][j] = lrelu(s);
  }
  __syncthreads();

  const int nc = lane & 15;
  const int mb = (lane >> 4) << 3;
  for (int n0 = 0; n0 < 64; n0 += 16) {
    v8f acc = wmma_tile64(tS, W_tog, 64, n0, lane);
    for (int r = 0; r < 8; ++r) {
      float v = acc[r] + b_tog[n0 + nc];
      x[(m0 + mb + r) * 64 + n0 + nc] = lrelu(v);
    }
  }
}

// ---------------------------------------------------------------------------
// Degree / dinv
// ---------------------------------------------------------------------------
__global__ void deg_init_kernel(float* __restrict__ deg, int N) {
  int i = blockIdx.x * blockDim.x + threadIdx.x;
  if (i < N) deg[i] = 1.0f;  // self-loop contribution
}

__global__ void deg_edges_kernel(const int* __restrict__ dst,
                                 float* __restrict__ deg, int E) {
  int e = blockIdx.x * blockDim.x + threadIdx.x;
  if (e < E) atomicAdd(&deg[dst[e]], 1.0f);
}

__global__ void deg_finish_kernel(float* __restrict__ deg, int N) {
  int i = blockIdx.x * blockDim.x + threadIdx.x;
  if (i < N) {
    float d = deg[i];
    deg[i] = (d > 0.0f) ? rsqrtf(d) : 0.0f;  // now holds dinv
  }
}

// ---------------------------------------------------------------------------
// FAConv pieces
// ---------------------------------------------------------------------------
__global__ void node_dots_kernel(const float* __restrict__ h,
                                 const float* __restrict__ att_l,
                                 const float* __restrict__ att_r,
                                 float* __restrict__ al, float* __restrict__ ar,
                                 int N) {
  int i = blockIdx.x * blockDim.x + threadIdx.x;
  if (i >= N) return;
  const float* row = h + i * 64;
  float sl = 0.0f, sr = 0.0f;
  for (int d = 0; d < 64; ++d) {
    float v = row[d];
    sl += v * att_l[d];
    sr += v * att_r[d];
  }
  al[i] = sl;
  ar[i] = sr;
}

__global__ void eps_init_kernel(const float* __restrict__ h0,
                                float* __restrict__ out, int n) {
  int i = blockIdx.x * blockDim.x + threadIdx.x;
  if (i < n) out[i] = EPSF * h0[i];
}

// one wave per edge: 32 lanes, 2 channels each -> 128B-coalesced L2 atomics
__global__ void __launch_bounds__(256)
edge_scatter_kernel(const float* __restrict__ h, const float* __restrict__ al,
                    const float* __restrict__ ar,
                    const float* __restrict__ dinv,
                    const int* __restrict__ src, const int* __restrict__ dst,
                    float* __restrict__ out, int E) {
  int t = blockIdx.x * blockDim.x + threadIdx.x;
  int e = t >> 5;
  int lane = t & 31;
  if (e >= E) return;
  int s = src[e];
  int d = dst[e];
  float alpha = tanhf(al[s] + ar[d]) * dinv[s] * dinv[d];
  const float* hr = h + s * 64;
  float* orow = out + d * 64;
  atomicAdd(&orow[lane], hr[lane] * alpha);
  atomicAdd(&orow[lane + 32], hr[lane + 32] * alpha);
}

__global__ void self_loop_kernel(const float* __restrict__ h,
                                 const float* __restrict__ al,
                                 const float* __restrict__ ar,
                                 const float* __restrict__ dinv,
                                 float* __restrict__ out, int N) {
  int t = blockIdx.x * blockDim.x + threadIdx.x;
  int i = t >> 6;
  int d = t & 63;
  if (i >= N) return;
  float dv = dinv[i];
  float alpha = tanhf(al[i] + ar[i]) * dv * dv;
  out[i * 64 + d] += h[i * 64 + d] * alpha;
}

__global__ void smooth_kernel(float* __restrict__ v, int n) {
  int i = blockIdx.x * blockDim.x + threadIdx.x;
  if (i < n) {
    float a = v[i];
    v[i] = sqrtf(a * a + 1e-8f);
  }
}

// ---------------------------------------------------------------------------
// CSR per-user pool: block-per-user, 64 lanes = 64 channels.
// Matches searchsorted(side='right')-1 on sorted offsets.
// ---------------------------------------------------------------------------
__global__ void __launch_bounds__(64)
user_pool_kernel(const float* __restrict__ x2, const int* __restrict__ offs,
                 float* __restrict__ x3) {
  int u = blockIdx.x;
  int d = threadIdx.x;
  int b = offs[u];
  int e = offs[u + 1];
  float s = 0.0f;
  for (int i = b; i < e; ++i) s += x2[i * 64 + d];
  x3[u * 64 + d] = s;
}

// ---------------------------------------------------------------------------
// Final user MLP with gather: one wave per 16 users.
// ---------------------------------------------------------------------------
__global__ void __launch_bounds__(32)
user_mlp_kernel(const float* __restrict__ x3, const int* __restrict__ re_index,
                const float* __restrict__ W_f1,   // [64,32]
                const float* __restrict__ b_f1,   // [32]
                const float* __restrict__ W_lab,  // [32,2]
                const float* __restrict__ b_lab,  // [2]
                float* __restrict__ out, int U) {
  __shared__ float gS[16][68];
  __shared__ float hS[16][36];
  const int lane = threadIdx.x;
  const int u0 = blockIdx.x * 16;

  for (int e = lane; e < 16 * 64; e += 32) {
    const int m = e >> 6, j = e & 63;
    const int srcu = re_index[u0 + m];
    gS[m][j] = x3[srcu * 64 + j];
  }
  __syncthreads();

  const int nc = lane & 15;
  const int mb = (lane >> 4) << 3;
  for (int n0 = 0; n0 < 32; n0 += 16) {
    v8f acc = wmma_tile64(gS, W_f1, 32, n0, lane);
    for (int r = 0; r < 8; ++r) {
      float v = acc[r] + b_f1[n0 + nc];
      hS[mb + r][n0 + nc] = lrelu(v);
    }
  }
  __syncthreads();

  // 32 lanes = 16 users x 2 output classes
  const int m = lane >> 1;
  const int c = lane & 1;
  float s = b_lab[c];
  for (int j = 0; j < 32; ++j) s += hS[m][j] * W_lab[j * 2 + c];
  out[(u0 + m) * 2 + c] = s;
}

// ---------------------------------------------------------------------------
// Host-side launch
// ---------------------------------------------------------------------------
extern "C" void kernel_launch(void* const* d_in, const int* in_sizes, int n_in,
                              void* d_out, int out_size, void* d_ws,
                              size_t ws_size, hipStream_t stream) {
  const int N = in_sizes[0] / 20;  // 200000
  const int E = in_sizes[3] / 2;   // 1600000
  const int U = in_sizes[4];       // 20000

  const float* num_prop = (const float*)d_in[0];
  const float* cat_prop = (const float*)d_in[1];
  const int* offs       = (const int*)d_in[2];
  const int* edge       = (const int*)d_in[3];  // [2,E]: src=edge, dst=edge+E
  const int* re_index   = (const int*)d_in[4];
  const float* W_num = (const float*)d_in[5];
  const float* b_num = (const float*)d_in[6];
  const float* W_cat = (const float*)d_in[7];
  const float* b_cat = (const float*)d_in[8];
  const float* W_tog = (const float*)d_in[9];
  const float* b_tog = (const float*)d_in[10];
  const float* att_l = (const float*)d_in[11];
  const float* att_r = (const float*)d_in[12];
  const float* W_f1  = (const float*)d_in[13];
  const float* b_f1  = (const float*)d_in[14];
  const float* W_lab = (const float*)d_in[15];
  const float* b_lab = (const float*)d_in[16];
  float* out = (float*)d_out;

  // workspace layout (all fp32)
  float* ws = (float*)d_ws;
  float* x    = ws;                 // [N,64]
  float* x1   = x + (size_t)N * 64; // [N,64]
  float* x2   = x1 + (size_t)N * 64;// [N,64]
  float* al   = x2 + (size_t)N * 64;// [N]
  float* ar   = al + N;             // [N]
  float* dinv = ar + N;             // [N]  (also deg scratch)
  float* x3   = dinv + N;           // [U,64]

  const int NE = N * 64;
  const int src_off = 0, dst_off = E;

  // 1) node features -> x
  node_mlp_kernel<<<N / 16, 32, 0, stream>>>(num_prop, cat_prop, W_num, b_num,
                                             W_cat, b_cat, W_tog, b_tog, x, N);

  // 2) gcn_norm degrees -> dinv
  deg_init_kernel<<<(N + 255) / 256, 256, 0, stream>>>(dinv, N);
  deg_edges_kernel<<<(E + 255) / 256, 256, 0, stream>>>(edge + dst_off, dinv, E);
  deg_finish_kernel<<<(N + 255) / 256, 256, 0, stream>>>(dinv, N);

  // 3) FAConv layer 1: x -> x1 (h0 = x)
  node_dots_kernel<<<(N + 255) / 256, 256, 0, stream>>>(x, att_l, att_r, al, ar, N);
  eps_init_kernel<<<(NE + 255) / 256, 256, 0, stream>>>(x, x1, NE);
  edge_scatter_kernel<<<(E * 32 + 255) / 256, 256, 0, stream>>>(
      x, al, ar, dinv, edge + src_off, edge + dst_off, x1, E);
  self_loop_kernel<<<(NE + 255) / 256, 256, 0, stream>>>(x, al, ar, dinv, x1, N);

  // 4) FAConv layer 2: x1 -> x2 (h0 = x)
  node_dots_kernel<<<(N + 255) / 256, 256, 0, stream>>>(x1, att_l, att_r, al, ar, N);
  eps_init_kernel<<<(NE + 255) / 256, 256, 0, stream>>>(x, x2, NE);
  edge_scatter_kernel<<<(E * 32 + 255) / 256, 256, 0, stream>>>(
      x1, al, ar, dinv, edge + src_off, edge + dst_off, x2, E);
  self_loop_kernel<<<(NE + 255) / 256, 256, 0, stream>>>(x1, al, ar, dinv, x2, N);

  // 5) smooth: x2 = sqrt(x2^2 + 1e-8)
  smooth_kernel<<<(NE + 255) / 256, 256, 0, stream>>>(x2, NE);

  // 6) per-user CSR pool -> x3
  user_pool_kernel<<<U, 64, 0, stream>>>(x2, offs, x3);

  // 7) gathered user MLP -> out[U,2]
  user_mlp_kernel<<<U / 16, 32, 0, stream>>>(x3, re_index, W_f1, b_f1, W_lab,
                                             b_lab, out, U);
}